// EulerEquationModel_3023656976488
// MI455X (gfx1250) — compile-verified
//
#include <hip/hip_runtime.h>

typedef __attribute__((ext_vector_type(2))) float v2f;
typedef __attribute__((ext_vector_type(8))) float v8f;

#define IX     1440
#define IY     720
#define PLANE  (IY * IX)            // 1,036,800
#define CH     12
#define CHPLANE (CH * PLANE)        // 12,441,600
#define NB     4
#define COLS   (NB * PLANE)         // 4,147,200 columns (b,y,x)
#define NFIELD ((long long)NB * CHPLANE)  // 49,766,400 output field elems

// ---------------------------------------------------------------------------
// Main fused kernel: WMMA channel-mix einsum + circular stencils + pointwise
// combine + per-block partial reductions.
// One thread = one (b,y,x) column (all 12 channels). One wave = 32 columns,
// einsum done as 2 pixel-groups x 3 chained V_WMMA_F32_16X16X4_F32.
// ---------------------------------------------------------------------------
__global__ __launch_bounds__(256) void euler_main(
    const float* __restrict__ F,      // (4,12,720,1440)
    const float* __restrict__ W,      // (15,12) row-major
    const float* __restrict__ bias,   // (15,)
    const float* __restrict__ tf_p,   // (1,)
    float* __restrict__ out,          // field (49,766,400) then 4 scalars
    float* __restrict__ ws)           // per-block partials (4 floats/block)
{
    // 20 floats per pixel slot: 16B-aligned float4 access, LDS-bank friendly
    __shared__ float mlbuf[256 * 20];   // 20 KB

    const int tid  = threadIdx.x;
    const int lane = tid & 31;
    const int wave = tid >> 5;
    const bool lo  = lane < 16;
    const int row  = lane & 15;
    const int qBlock = (int)blockIdx.x * 256;

    // ---- A operand: W matrix, M=out-channel (row), K split lanes 0-15/16-31
    v2f a[3];
#pragma unroll
    for (int k = 0; k < 3; ++k) {
        const int c0 = 4 * k + (lo ? 0 : 2);
        v2f av;
        av[0] = (row < 15) ? W[row * 12 + c0]     : 0.0f;
        av[1] = (row < 15) ? W[row * 12 + c0 + 1] : 0.0f;
        a[k] = av;
    }
    // ---- C init = bias broadcast (handles "+ b_bias" for free)
    v8f cinit;
#pragma unroll
    for (int m = 0; m < 8; ++m) {
        const int oc = lo ? m : (m + 8);
        cinit[m] = (oc < 15) ? bias[oc] : 0.0f;
    }

    // ---- einsum over the wave's 32 columns: two 16-pixel groups
#pragma unroll
    for (int g = 0; g < 2; ++g) {
        const unsigned q = (unsigned)(qBlock + wave * 32 + g * 16 + row);
        const unsigned bb  = q / (unsigned)PLANE;
        const unsigned rem = q - bb * (unsigned)PLANE;
        const unsigned yy  = rem / (unsigned)IX;
        const unsigned xx  = rem - yy * (unsigned)IX;
        const long long colbase =
            (long long)bb * CHPLANE + (long long)yy * IX + xx;

        v8f acc = cinit;
#pragma unroll
        for (int k = 0; k < 3; ++k) {
            const int c0 = 4 * k + (lo ? 0 : 2);
            v2f bv;
            bv[0] = F[colbase + (long long)c0 * PLANE];
            bv[1] = F[colbase + (long long)(c0 + 1) * PLANE];
            acc = __builtin_amdgcn_wmma_f32_16x16x4_f32(
                false, a[k], false, bv, (short)0, acc, false, false);
        }
        // lane n<16 -> channels 0..7 of pixel n; lane n+16 -> channels 8..15
        const int pix = wave * 32 + g * 16 + row;
        float* dst = &mlbuf[pix * 20 + (lo ? 0 : 8)];
        ((float4*)dst)[0] = make_float4(acc[0], acc[1], acc[2], acc[3]);
        ((float4*)dst)[1] = make_float4(acc[4], acc[5], acc[6], acc[7]);
    }
    __syncthreads();

    // ---- per-column stencil + combine phase ----
    const unsigned q = (unsigned)(qBlock + tid);
    const unsigned bb  = q / (unsigned)PLANE;
    const unsigned rem = q - bb * (unsigned)PLANE;
    const int y = (int)(rem / (unsigned)IX);
    const int x = (int)(rem - (unsigned)y * (unsigned)IX);

    const float* Fb = F   + (long long)bb * CHPLANE;
    float*       Ob = out + (long long)bb * CHPLANE;

    const int xc  = y * IX + x;
    const int xm1 = y * IX + (x == 0      ? IX - 1 : x - 1);
    const int xm2 = y * IX + (x <= 1      ? x + IX - 2 : x - 2);
    const int xp1 = y * IX + (x == IX - 1 ? 0 : x + 1);
    const int xp2 = y * IX + (x >= IX - 2 ? x - (IX - 2) : x + 2);
    const int ym1 = (y == 0      ? IY - 1 : y - 1) * IX + x;
    const int ym2 = (y <= 1      ? y + IY - 2 : y - 2) * IX + x;
    const int yp1 = (y == IY - 1 ? 0 : y + 1) * IX + x;
    const int yp2 = (y >= IY - 2 ? y - (IY - 2) : y + 2) * IX + x;

    // pull this column's 15 ml values back from LDS (float4, 16B aligned)
    float ml[16];
    {
        const float4* src = (const float4*)&mlbuf[tid * 20];
        float4 m0 = src[0], m1 = src[1], m2 = src[2], m3 = src[3];
        ml[0] = m0.x; ml[1] = m0.y; ml[2] = m0.z; ml[3] = m0.w;
        ml[4] = m1.x; ml[5] = m1.y; ml[6] = m1.z; ml[7] = m1.w;
        ml[8] = m2.x; ml[9] = m2.y; ml[10] = m2.z; ml[11] = m2.w;
        ml[12] = m3.x; ml[13] = m3.y; ml[14] = m3.z; ml[15] = m3.w;
    }

    float ctr[12];
#pragma unroll
    for (int c = 0; c < CH; ++c) ctr[c] = Fb[(long long)c * PLANE + xc];

    const float tf = tf_p[0];
    const float w1 = 1.0f / 12.0f, w8 = 8.0f / 12.0f;
    const float plist[3] = {10.0f, 8.5f, 5.0f};

    float accCon = 0.0f, accEF = 0.0f, accPP = 0.0f, accXY = 0.0f;

#pragma unroll
    for (int z = 0; z < 3; ++z) {
        const int zp = (z + 1) % 3;
        const int zm = (z + 2) % 3;
        float fdx[4], fdy[4], fdz[4];
#pragma unroll
        for (int s = 0; s < 4; ++s) {
            const int c = s * 3 + z;
            const float* P = Fb + (long long)c * PLANE;
            const float dxm2 = P[xm2], dxm1 = P[xm1];
            const float dxp1 = P[xp1], dxp2 = P[xp2];
            const float dym2 = P[ym2], dym1 = P[ym1];
            const float dyp1 = P[yp1], dyp2 = P[yp2];
            fdx[s] = w1 * dxm2 - w8 * dxm1 + w8 * dxp1 - w1 * dxp2;
            fdy[s] = w1 * dym2 - w8 * dym1 + w8 * dyp1 - w1 * dyp2;
            fdz[s] = 0.5f * (ctr[s * 3 + zp] - ctr[s * 3 + zm]);
        }
        const float o = ml[12 + z];
        const float u = ctr[z], v = ctr[3 + z], T = ctr[6 + z];
        const float term3 = tf * T / plist[z] * o;
#pragma unroll
        for (int s = 0; s < 4; ++s) {
            const float ef  = ml[s * 3 + z];
            const float xyd = -u * fdx[s] - v * fdy[s];
            const float phys = (s == 0) ? -fdx[3]
                             : (s == 1) ? -fdy[3]
                             : (s == 2) ? term3 : 0.0f;
            const float pp = xyd - o * fdz[s] + phys;
            Ob[(long long)(s * 3 + z) * PLANE + xc] = ctr[s * 3 + z] + ef + pp;
            accEF += ef * ef; accPP += pp * pp; accXY += xyd * xyd;
        }
        const float dzo = 0.5f * (ml[12 + zp] - ml[12 + zm]);
        const float con = fdx[0] + fdy[1] + dzo;
        accCon += con * con;
    }

    // ---- deterministic per-block reduction (reuse LDS) ----
    __syncthreads();
    float* red = mlbuf;   // need 1024 floats, have 5120
    red[tid]       = accCon;
    red[256 + tid] = accEF;
    red[512 + tid] = accPP;
    red[768 + tid] = accXY;
    __syncthreads();
#pragma unroll
    for (int st = 128; st > 0; st >>= 1) {
        if (tid < st) {
            red[tid]       += red[tid + st];
            red[256 + tid] += red[256 + tid + st];
            red[512 + tid] += red[512 + tid + st];
            red[768 + tid] += red[768 + tid + st];
        }
        __syncthreads();
    }
    if (tid == 0) {
        ((float4*)ws)[blockIdx.x] =
            make_float4(red[0], red[256], red[512], red[768]);
    }
}

// ---------------------------------------------------------------------------
// Final reduction: single block, fixed summation order -> deterministic.
// ---------------------------------------------------------------------------
__global__ __launch_bounds__(256) void euler_reduce(
    const float* __restrict__ ws, float* __restrict__ out, int nBlocks)
{
    __shared__ float red[1024];
    const int tid = threadIdx.x;
    float a0 = 0.f, a1 = 0.f, a2 = 0.f, a3 = 0.f;
    for (int i = tid; i < nBlocks; i += 256) {
        const float4 w4 = ((const float4*)ws)[i];
        a0 += w4.x; a1 += w4.y; a2 += w4.z; a3 += w4.w;
    }
    red[tid] = a0; red[256 + tid] = a1; red[512 + tid] = a2; red[768 + tid] = a3;
    __syncthreads();
#pragma unroll
    for (int st = 128; st > 0; st >>= 1) {
        if (tid < st) {
            red[tid]       += red[tid + st];
            red[256 + tid] += red[256 + tid + st];
            red[512 + tid] += red[512 + tid + st];
            red[768 + tid] += red[768 + tid + st];
        }
        __syncthreads();
    }
    if (tid == 0) {
        const float invCon = 1.0f / (float)CHPLANE;       // 4*3*720*1440 elems... = 12,441,600
        const float invAll = 1.0f / (float)NFIELD;        // 49,766,400 elems
        out[NFIELD + 0] = red[0]   * invCon;
        out[NFIELD + 1] = red[256] * invAll;
        out[NFIELD + 2] = red[512] * invAll;
        out[NFIELD + 3] = red[768] * invAll;
    }
}

extern "C" void kernel_launch(void* const* d_in, const int* in_sizes, int n_in,
                              void* d_out, int out_size, void* d_ws, size_t ws_size,
                              hipStream_t stream)
{
    (void)in_sizes; (void)n_in; (void)out_size; (void)ws_size;
    const float* F    = (const float*)d_in[0];
    const float* W    = (const float*)d_in[1];
    const float* bias = (const float*)d_in[2];
    const float* tf   = (const float*)d_in[3];
    float* out = (float*)d_out;
    float* ws  = (float*)d_ws;

    const int nBlocks = COLS / 256;   // 16,200
    euler_main<<<nBlocks, 256, 0, stream>>>(F, W, bias, tf, out, ws);
    euler_reduce<<<1, 256, 0, stream>>>(ws, out, nBlocks);
}